// ZernikeKernelConv_65309272703463
// MI455X (gfx1250) — compile-verified
//
#include <hip/hip_runtime.h>

#define WPB 2  // waves per block (wave32)

typedef __attribute__((ext_vector_type(2))) float v2f;
typedef __attribute__((ext_vector_type(8))) float v8f;

// ---------------- compile-time tables (monomial basis, offset matrix) -------
struct Tab {
  int   ex[20][3];
  float oc[20][20][2];  // [0]=coef, [1]=(float)oidx
};

constexpr int combi(int n, int k) {
  int r = 1;
  for (int t = 0; t < k; ++t) r = r * (n - t) / (t + 1);
  return r;
}

constexpr Tab make_tab() {
  Tab t{};
  int ex[20][3] = {};
  int cnt = 0;
  for (int i = 0; i <= 3; ++i)
    for (int j = 0; j <= 3; ++j)
      for (int k = 0; k <= 3; ++k)
        if (i + j + k <= 3) { ex[cnt][0] = i; ex[cnt][1] = j; ex[cnt][2] = k; ++cnt; }
  for (int q = 0; q < 20; ++q) {
    t.ex[q][0] = ex[q][0]; t.ex[q][1] = ex[q][1]; t.ex[q][2] = ex[q][2];
  }
  for (int r = 0; r < 20; ++r)
    for (int s = 0; s < 20; ++s) {
      int di = ex[r][0] - ex[s][0];
      int dj = ex[r][1] - ex[s][1];
      int dk = ex[r][2] - ex[s][2];
      if (di >= 0 && dj >= 0 && dk >= 0) {
        int m = 0;
        for (int q = 0; q < 20; ++q)
          if (ex[q][0] == di && ex[q][1] == dj && ex[q][2] == dk) m = q;
        float c = (float)(combi(ex[r][0], ex[s][0]) * combi(ex[r][1], ex[s][1]) *
                          combi(ex[r][2], ex[s][2]));
        if ((di + dj + dk) & 1) c = -c;
        t.oc[r][s][0] = c;
        t.oc[r][s][1] = (float)m;
      } else {
        t.oc[r][s][0] = 0.0f;
        t.oc[r][s][1] = 0.0f;
      }
    }
  return t;
}

constexpr Tab TAB = make_tab();

// ---------------- helpers ---------------------------------------------------
__device__ __forceinline__ float ipow(float v, int e) {
  float r = 1.0f;
  for (int t = 0; t < e; ++t) r *= v;
  return r;
}

// D(16x16 f32) = A(16x4 f32) * B(4x16 f32) + C
__device__ __forceinline__ v8f wmma4(v2f a, v2f b, v8f c) {
  return __builtin_amdgcn_wmma_f32_16x16x4_f32(false, a, false, b, (short)0, c,
                                               false, false);
}

// ---------------- pre-reduction: rnpb[b] = 1 / mean(patches_size[b,:]) ------
__global__ void npb_kernel(const float* __restrict__ psize, float* __restrict__ rnpb, int N) {
  __shared__ float red[256];
  const int b = blockIdx.x;
  float s = 0.0f;
  for (int i = threadIdx.x; i < N; i += blockDim.x) s += psize[(long long)b * N + i];
  red[threadIdx.x] = s;
  __syncthreads();
  for (int st = 128; st > 0; st >>= 1) {
    if ((int)threadIdx.x < st) red[threadIdx.x] += red[threadIdx.x + st];
    __syncthreads();
  }
  if (threadIdx.x == 0) rnpb[b] = (float)N / red[0];
}

// ---------------- main kernel: one wave == one point ------------------------
// B-side LDS buffers use a row-pair-packed layout: element (row,col) lives at
// v2f slot [(row>>1)*NC + col], .x = even row, .y = odd row.  A WMMA B operand
// {B[k][n], B[k+1][n]} is then ONE aligned ds_load_b64 (k always even here).
__global__ __launch_bounds__(WPB * 32) void zern_wmma_kernel(
    const float* __restrict__ points, const float* __restrict__ radius,
    const float* __restrict__ feat0, const int* __restrict__ patches_idx,
    const float* __restrict__ patches_dist,
    const float* __restrict__ Z0, const float* __restrict__ Z1,
    const float* __restrict__ Z2, const float* __restrict__ Z3,
    const float* __restrict__ rnpbp, float* __restrict__ out, int B, int N) {
  __shared__ __align__(16) float sZ[32 * 32];          // Zall (A layout, row-major)
  __shared__ __align__(16) float sA[WPB][32 * 32];     // sg^T (A layout, row-major)
  __shared__ __align__(16) float sF[WPB][32 * 64];     // fg   (pair-packed B)
  __shared__ __align__(16) float sY[WPB][32 * 64];     // yl   (pair-packed B)
  __shared__ __align__(16) float sO[WPB][32 * 32];     // om   (pair-packed B)
  __shared__ __align__(16) float sM[WPB][32 * 32];     // Mtot (A layout, row-major)
  __shared__ __align__(16) float sT[WPB][32];          // target monomials

  const int tid = threadIdx.x;

  // ---- stage Zall (rows: Z0 flat 0..1, Z1 2..7, Z2 8..12, Z3 13..19) ----
  for (int e = tid; e < 32 * 32; e += blockDim.x) {
    const int g = e >> 5, j = e & 31;
    float v = 0.0f;
    if (g < 20 && j < 20) {
      if (g < 2)       v = Z0[g * 20 + j];
      else if (g < 8)  v = Z1[(g - 2) * 20 + j];
      else if (g < 13) v = Z2[(g - 8) * 20 + j];
      else             v = Z3[(g - 13) * 20 + j];
    }
    sZ[e] = v;
  }
  __syncthreads();

  const int w = tid >> 5, lane = tid & 31;
  const float rinv = 1.0f / radius[0];   // hoisted reciprocal (once per wave)
  const long long BN = (long long)B * N;
  const long long waveId = (long long)blockIdx.x * WPB + w;
  const long long nWaves = (long long)gridDim.x * WPB;
  const int m16 = lane & 15;          // col (B/D) or row (A)
  const int hi  = lane >> 4;          // half-wave id
  const int kk2 = hi << 1;            // K pair base within 4-step (A side)
  const int m08 = hi << 3;            // D row base

  float* mysA = sA[w];
  float* mysM = sM[w];
  float* mysT = sT[w];
  v2f* pF = (v2f*)sF[w];
  v2f* pY = (v2f*)sY[w];
  v2f* pO = (v2f*)sO[w];

  const v8f vzero = {0, 0, 0, 0, 0, 0, 0, 0};

  for (long long pt = waveId; pt < BN; pt += nWaves) {
    const int b = (int)(pt / N);
    const float rnpb = rnpbp[b];

    if (pt + nWaves < BN) __builtin_prefetch(patches_idx + (pt + nWaves) * 32, 0, 1);

    // ---- build A = sg^T : lane handles neighbor p = lane ----
    const int p = lane;
    const int nb = patches_idx[pt * 32 + p];
    const float dist = patches_dist[pt * 32 + p];
    const float msk = (dist <= 1.0f) ? 1.0f : 0.0f;
    const float* nptr = points + ((long long)b * N + nb) * 3;
    const float nx = nptr[0] * rinv, ny = nptr[1] * rinv, nz = nptr[2] * rinv;
    {
      const float xs[4] = {1.0f, nx, nx * nx, nx * nx * nx};
      const float ys[4] = {1.0f, ny, ny * ny, ny * ny * ny};
      const float zs[4] = {1.0f, nz, nz * nz, nz * nz * nz};
#pragma unroll
      for (int k = 0; k < 20; ++k)
        mysA[k * 32 + p] = msk * (xs[TAB.ex[k][0]] * ys[TAB.ex[k][1]] * zs[TAB.ex[k][2]]);
      // rows 20..31 are only consumed by yl rows 20..31, which are never read
      // downstream (matmul3 k-steps stop at row 19) -> no zero-fill needed.
    }

    // ---- stage fg (32 x 64) pair-packed: two neighbor rows interleaved ----
#pragma unroll
    for (int it = 0; it < 8; ++it) {
      const int e = it * 32 + lane;   // 256 tasks = 16 row-pairs x 16 col-quads
      const int pr = e >> 4;          // row pair 0..15
      const int qd = e & 15;          // col quad 0..15
      const int n0 = __shfl(nb, pr * 2);
      const int n1 = __shfl(nb, pr * 2 + 1);
      const float4 a4 = ((const float4*)(feat0 + ((long long)b * N + n0) * 64))[qd];
      const float4 b4 = ((const float4*)(feat0 + ((long long)b * N + n1) * 64))[qd];
      const int base = pr * 64 + qd * 4;
      v2f t0; t0.x = a4.x; t0.y = b4.x;
      v2f t1; t1.x = a4.y; t1.y = b4.y;
      v2f t2; t2.x = a4.z; t2.y = b4.z;
      v2f t3; t3.x = a4.w; t3.y = b4.w;
      pF[base + 0] = t0;
      pF[base + 1] = t1;
      pF[base + 2] = t2;
      pF[base + 3] = t3;
    }

    // ---- target monomials + om matrix (pair-packed, two rows at a time) ----
    {
      const float* tptr = points + pt * 3;
      const float px = tptr[0] * rinv, py = tptr[1] * rinv, pz = tptr[2] * rinv;
      float tv = 0.0f;
      if (lane < 20)
        tv = ipow(px, TAB.ex[lane][0]) * ipow(py, TAB.ex[lane][1]) *
             ipow(pz, TAB.ex[lane][2]);
      mysT[lane] = tv;
#pragma unroll
      for (int r2 = 0; r2 < 10; ++r2) {
        float o0 = 0.0f, o1 = 0.0f;
        if (lane < 20) {
          o0 = mysT[(int)TAB.oc[2 * r2][lane][1]] * TAB.oc[2 * r2][lane][0];
          o1 = mysT[(int)TAB.oc[2 * r2 + 1][lane][1]] * TAB.oc[2 * r2 + 1][lane][0];
        }
        v2f t; t.x = o0; t.y = o1;
        pO[r2 * 32 + lane] = t;       // cols 20..31 get zeros
      }
    }

    // ---- matmul1: yl(20x64) = sg^T(20x32) x fg(32x64), scale 1/npb ----
    // two c-halves of 2 tiles each to keep accumulator pressure at 32 VGPRs
#pragma unroll 1
    for (int ch = 0; ch < 2; ++ch) {
      v8f acc[2][2];
#pragma unroll
      for (int mt = 0; mt < 2; ++mt)
#pragma unroll
        for (int c2 = 0; c2 < 2; ++c2) acc[mt][c2] = vzero;
#pragma unroll
      for (int ks = 0; ks < 8; ++ks) {
        const v2f a0 = *(const v2f*)(mysA + (0 + m16) * 32 + ks * 4 + kk2);
        const v2f a1 = *(const v2f*)(mysA + (16 + m16) * 32 + ks * 4 + kk2);
#pragma unroll
        for (int c2 = 0; c2 < 2; ++c2) {
          const int cb = (ch * 2 + c2) * 16;
          const v2f bb = pF[(ks * 2 + hi) * 64 + cb + m16];
          acc[0][c2] = wmma4(a0, bb, acc[0][c2]);
          acc[1][c2] = wmma4(a1, bb, acc[1][c2]);
        }
      }
      // store yl pair-packed straight from accumulators (i, i+1 = row pair)
#pragma unroll
      for (int mt = 0; mt < 2; ++mt)
#pragma unroll
        for (int c2 = 0; c2 < 2; ++c2)
#pragma unroll
          for (int i2 = 0; i2 < 4; ++i2) {
            v2f t;
            t.x = acc[mt][c2][2 * i2] * rnpb;
            t.y = acc[mt][c2][2 * i2 + 1] * rnpb;
            pY[(mt * 8 + hi * 4 + i2) * 64 + (ch * 2 + c2) * 16 + m16] = t;
          }
    }

    // ---- matmul2: Mtot(20x20) = Zall(20x20) x om(20x20) ----
    {
      v8f m2[2][2];
#pragma unroll
      for (int mt = 0; mt < 2; ++mt)
#pragma unroll
        for (int nt = 0; nt < 2; ++nt) m2[mt][nt] = vzero;
#pragma unroll
      for (int ks = 0; ks < 5; ++ks) {
        const v2f a0 = *(const v2f*)(sZ + (0 + m16) * 32 + ks * 4 + kk2);
        const v2f a1 = *(const v2f*)(sZ + (16 + m16) * 32 + ks * 4 + kk2);
#pragma unroll
        for (int nt = 0; nt < 2; ++nt) {
          const v2f bb = pO[(ks * 2 + hi) * 32 + nt * 16 + m16];
          m2[0][nt] = wmma4(a0, bb, m2[0][nt]);
          m2[1][nt] = wmma4(a1, bb, m2[1][nt]);
        }
      }
#pragma unroll
      for (int mt = 0; mt < 2; ++mt)
#pragma unroll
        for (int nt = 0; nt < 2; ++nt)
#pragma unroll
          for (int i = 0; i < 8; ++i)
            mysM[(mt * 16 + m08 + i) * 32 + nt * 16 + m16] = m2[mt][nt][i];
    }

    // ---- matmul3: res(20x64) = Mtot(20x20) x yl(20x64), two c-halves ----
#pragma unroll 1
    for (int ch = 0; ch < 2; ++ch) {
      v8f acc[2][2];
#pragma unroll
      for (int mt = 0; mt < 2; ++mt)
#pragma unroll
        for (int c2 = 0; c2 < 2; ++c2) acc[mt][c2] = vzero;
#pragma unroll
      for (int ks = 0; ks < 5; ++ks) {
        const v2f a0 = *(const v2f*)(mysM + (0 + m16) * 32 + ks * 4 + kk2);
        const v2f a1 = *(const v2f*)(mysM + (16 + m16) * 32 + ks * 4 + kk2);
#pragma unroll
        for (int c2 = 0; c2 < 2; ++c2) {
          const int cb = (ch * 2 + c2) * 16;
          const v2f bb = pY[(ks * 2 + hi) * 64 + cb + m16];
          acc[0][c2] = wmma4(a0, bb, acc[0][c2]);
          acc[1][c2] = wmma4(a1, bb, acc[1][c2]);
        }
      }

      // ---- scatter rows g=0..19 of this c-half into output segments ----
#pragma unroll
      for (int mt = 0; mt < 2; ++mt) {
#pragma unroll
        for (int i = 0; i < 8; ++i) {
          const int g = mt * 16 + m08 + i;
          if (g < 20) {
            int rl, rb;
            long long soff;
            if (g < 2)       { rl = 2; rb = 0;  soff = 0; }
            else if (g < 8)  { rl = 6; rb = 2;  soff = BN * 128; }
            else if (g < 13) { rl = 5; rb = 8;  soff = BN * 512; }
            else             { rl = 7; rb = 13; soff = BN * 832; }
            const long long base = soff + ((long long)pt * rl + (g - rb)) * 64;
#pragma unroll
            for (int c2 = 0; c2 < 2; ++c2)
              out[base + (ch * 2 + c2) * 16 + m16] = acc[mt][c2][i];
          }
        }
      }
    }
  }
}

// ---------------- launcher --------------------------------------------------
extern "C" void kernel_launch(void* const* d_in, const int* in_sizes, int n_in,
                              void* d_out, int out_size, void* d_ws, size_t ws_size,
                              hipStream_t stream) {
  const float* points = (const float*)d_in[0];
  const float* radius = (const float*)d_in[1];
  const float* feat0  = (const float*)d_in[2];
  const int*   pidx   = (const int*)d_in[3];
  const float* pdist  = (const float*)d_in[4];
  const float* psize  = (const float*)d_in[5];
  const float* Z0     = (const float*)d_in[6];
  const float* Z1     = (const float*)d_in[7];
  const float* Z2     = (const float*)d_in[8];
  const float* Z3     = (const float*)d_in[9];
  float* out = (float*)d_out;
  float* rnpb = (float*)d_ws;

  const int B = 2;
  const long long BN = (long long)in_sizes[5];  // patches_size is [B,N]
  const int N = (int)(BN / B);

  npb_kernel<<<B, 256, 0, stream>>>(psize, rnpb, N);

  const int ptsPerWave = 8;
  long long waves = (BN + ptsPerWave - 1) / ptsPerWave;
  int blocks = (int)((waves + WPB - 1) / WPB);
  if (blocks < 1) blocks = 1;
  zern_wmma_kernel<<<blocks, WPB * 32, 0, stream>>>(points, radius, feat0, pidx,
                                                    pdist, Z0, Z1, Z2, Z3, rnpb,
                                                    out, B, N);
}